// WSDModel_48077863911562
// MI455X (gfx1250) — compile-verified
//
#include <hip/hip_runtime.h>
#include <hip/hip_bf16.h>

// ---------------------------------------------------------------------------
// Problem constants (from reference): B=8, N=512, D=300, Y=30000
// ---------------------------------------------------------------------------
#define SEQ   512
#define BATCH 8
#define DIM   300
#define DIMP  304              // D padded to multiple of 16
#define YDIM  30000
#define ROWS  (BATCH * SEQ)    // 4096
#define NEGF  (-10000.0f)

#define AS_STRIDE 20           // 128x16 A tile, padded: conflict-free frags
#define BS_STRIDE 136          // 16x128 B tile, padded: half-waves disjoint
#define AS_ELEMS  (128 * AS_STRIDE)
#define BS_ELEMS  (16 * BS_STRIDE)

typedef __attribute__((ext_vector_type(2))) float v2f;
typedef __attribute__((ext_vector_type(8))) float v8f;

// ---------------------------------------------------------------------------
// f32 WMMA GEMM:  C = A * B   (BT=false, B is K x N row-major)
//                 C = A * B^T (BT=true,  B is N x K row-major)
// Block tile 128x128, K-tile 16, 8 waves (4 along M x 2 along N); each wave
// computes 32x64 via 8 f32 16x16x4 accumulators (32 WMMA per K-tile).
// Double-buffered LDS with register staging: global loads for tile k+1 are
// issued before the compute phase of tile k; one barrier per K-tile.
// Bounds zero-filled on load and checked on store.
// ---------------------------------------------------------------------------
template <bool BT>
__global__ __launch_bounds__(256) void gemm_wmma_f32(
    const float* __restrict__ Ag, const float* __restrict__ Bg,
    float* __restrict__ Cg, int M, int Nsz, int K, int lda, int ldb, int ldc,
    long long strideA, long long strideB, long long strideC) {
  const int bz = blockIdx.z;
  Ag += (long long)bz * strideA;
  Bg += (long long)bz * strideB;
  Cg += (long long)bz * strideC;

  __shared__ float As[2 * AS_ELEMS];
  __shared__ float Bs[2 * BS_ELEMS];

  const int t = threadIdx.x;
  const int lane = t & 31;
  const int wave = t >> 5;
  const int waveM = wave & 3;    // 0..3 along M (32 rows each)
  const int waveN = wave >> 2;   // 0..1 along N (64 cols each)
  const int lrow = lane & 15;
  const int lhi = lane >> 4;     // selects K-pair on src / M+8 half on dst
  const int rowBase = blockIdx.y * 128;
  const int colBase = blockIdx.x * 128;

  float aReg[8], bReg[8];

  // --- global -> register staging (zero-filled out of range) ---
  auto loadA = [&](int k0) {
#pragma unroll
    for (int i = 0; i < 8; ++i) {
      int e = t + i * 256;               // 128x16 = 2048 elems
      int r = e >> 4, c = e & 15;
      int gr = rowBase + r, gk = k0 + c;
      float v = 0.0f;
      if (gr < M && gk < K) v = Ag[(long long)gr * lda + gk];
      aReg[i] = v;
    }
  };
  auto loadB = [&](int k0) {
#pragma unroll
    for (int i = 0; i < 8; ++i) {
      int e = t + i * 256;               // 16x128 = 2048 elems
      int kr = e >> 7, c = e & 127;
      int gk = k0 + kr, gc = colBase + c;
      float v = 0.0f;
      if (BT) {
        if (gc < Nsz && gk < K) v = Bg[(long long)gc * ldb + gk];
      } else {
        if (gk < K && gc < Nsz) v = Bg[(long long)gk * ldb + gc];
      }
      bReg[i] = v;
    }
  };
  // --- register -> LDS staging ---
  auto storeTile = [&](int p) {
    float* Ap = As + p * AS_ELEMS;
    float* Bp = Bs + p * BS_ELEMS;
#pragma unroll
    for (int i = 0; i < 8; ++i) {
      int e = t + i * 256;
      Ap[(e >> 4) * AS_STRIDE + (e & 15)] = aReg[i];
    }
#pragma unroll
    for (int i = 0; i < 8; ++i) {
      int e = t + i * 256;
      Bp[(e >> 7) * BS_STRIDE + (e & 127)] = bReg[i];
    }
  };

  v8f acc[2][4] = {};
  const int nk = (K + 15) >> 4;

  loadA(0);
  loadB(0);
  storeTile(0);
  __syncthreads();

  for (int kb = 0; kb < nk; ++kb) {
    const int p = kb & 1;
    const bool more = (kb + 1) < nk;
    if (more) {            // issue next tile's global loads before compute
      loadA((kb + 1) * 16);
      loadB((kb + 1) * 16);
    }
    const float* Ap = As + p * AS_ELEMS;
    const float* Bp = Bs + p * BS_ELEMS;
#pragma unroll
    for (int kk = 0; kk < 16; kk += 4) {
      const int kidx = kk + 2 * lhi;     // per-ISA 16x16x4 f32 layout
      v2f a[2], b[4];
#pragma unroll
      for (int sm = 0; sm < 2; ++sm) {
        const int r = waveM * 32 + sm * 16 + lrow;
        a[sm].x = Ap[r * AS_STRIDE + kidx];
        a[sm].y = Ap[r * AS_STRIDE + kidx + 1];
      }
#pragma unroll
      for (int sn = 0; sn < 4; ++sn) {
        const int c = waveN * 64 + sn * 16 + lrow;
        b[sn].x = Bp[kidx * BS_STRIDE + c];
        b[sn].y = Bp[(kidx + 1) * BS_STRIDE + c];
      }
#pragma unroll
      for (int sm = 0; sm < 2; ++sm)
#pragma unroll
        for (int sn = 0; sn < 4; ++sn)
          acc[sm][sn] = __builtin_amdgcn_wmma_f32_16x16x4_f32(
              false, a[sm], false, b[sn], (short)0, acc[sm][sn], false, false);
    }
    if (more) storeTile(p ^ 1);  // other buffer: safe while peers compute on p
    __syncthreads();
  }

  // --- store D (VGPR j -> row j + 8*lhi, col = lane%16) ---
#pragma unroll
  for (int sm = 0; sm < 2; ++sm) {
#pragma unroll
    for (int sn = 0; sn < 4; ++sn) {
      const int col = colBase + waveN * 64 + sn * 16 + lrow;
      if (col < Nsz) {
#pragma unroll
        for (int j = 0; j < 8; ++j) {
          const int row = rowBase + waveM * 32 + sm * 16 + j + 8 * lhi;
          if (row < M) Cg[(long long)row * ldc + col] = acc[sm][sn][j];
        }
      }
    }
  }
}

// ---------------------------------------------------------------------------
// X[row] = E_v[M_s[row]] with zero padding to DIMP
// ---------------------------------------------------------------------------
__global__ __launch_bounds__(128) void gather_embed(
    const int* __restrict__ Ms, const float* __restrict__ Ev,
    float* __restrict__ X) {
  const int row = blockIdx.x;
  const int idx = Ms[row];
  const float* src = Ev + (long long)idx * DIM;
  float* dst = X + (long long)row * DIMP;
  for (int c = threadIdx.x; c < DIMP; c += 128)
    dst[c] = (c < DIM) ? src[c] : 0.0f;
}

// ---------------------------------------------------------------------------
// Row-wise: pad-mask (==0 -> -1e4), causal add (-1e4 for m>n), softmax.
// One block per (b,n) row of 512 scores; 256 threads, 2 elems each.
// ---------------------------------------------------------------------------
__global__ __launch_bounds__(256) void mask_softmax(
    const float* __restrict__ Sraw, float* __restrict__ Aout) {
  const int row = blockIdx.x;     // b*512 + n
  const int n = row & (SEQ - 1);  // position in sequence
  const float* s = Sraw + (long long)row * SEQ;
  float* o = Aout + (long long)row * SEQ;
  __shared__ float red[256];
  const int t = threadIdx.x;

  float v[2];
#pragma unroll
  for (int i = 0; i < 2; ++i) {
    const int m = t + i * 256;
    float x = s[m];
    if (x == 0.0f) x = NEGF;  // pad mask (matches jnp.where(A==0, NEG, A))
    if (m > n) x += NEGF;     // causal mask is *added* after the where
    v[i] = x;
  }
  float mx = fmaxf(v[0], v[1]);
  red[t] = mx;
  __syncthreads();
  for (int st = 128; st > 0; st >>= 1) {
    if (t < st) red[t] = fmaxf(red[t], red[t + st]);
    __syncthreads();
  }
  mx = red[0];
  __syncthreads();
  const float e0 = __expf(v[0] - mx), e1 = __expf(v[1] - mx);
  red[t] = e0 + e1;
  __syncthreads();
  for (int st = 128; st > 0; st >>= 1) {
    if (t < st) red[t] += red[t + st];
    __syncthreads();
  }
  const float inv = 1.0f / red[0];
  o[t] = e0 * inv;
  o[t + 256] = e1 * inv;
}

// ---------------------------------------------------------------------------
// H = LayerNorm(Qc + X) over the 300 real columns; pad cols zeroed.
// ---------------------------------------------------------------------------
__global__ __launch_bounds__(128) void residual_layernorm(
    const float* __restrict__ Qc, const float* __restrict__ X,
    const float* __restrict__ gamma, const float* __restrict__ beta,
    float* __restrict__ H) {
  const int row = blockIdx.x;
  const int t = threadIdx.x;
  const float* qr = Qc + (long long)row * DIMP;
  const float* xr = X + (long long)row * DIMP;
  float* hr = H + (long long)row * DIMP;
  __shared__ float red[128];

  float x[3];
  float s = 0.0f;
#pragma unroll
  for (int i = 0; i < 3; ++i) {
    const int c = t + i * 128;
    float v = 0.0f;
    if (c < DIM) v = qr[c] + xr[c];
    x[i] = v;
    s += v;
  }
  red[t] = s;
  __syncthreads();
  for (int st = 64; st > 0; st >>= 1) {
    if (t < st) red[t] += red[t + st];
    __syncthreads();
  }
  const float mu = red[0] * (1.0f / DIM);
  __syncthreads();
  float sq = 0.0f;
#pragma unroll
  for (int i = 0; i < 3; ++i) {
    const int c = t + i * 128;
    if (c < DIM) {
      const float d = x[i] - mu;
      sq += d * d;
    }
  }
  red[t] = sq;
  __syncthreads();
  for (int st = 64; st > 0; st >>= 1) {
    if (t < st) red[t] += red[t + st];
    __syncthreads();
  }
  const float rstd = rsqrtf(red[0] * (1.0f / DIM) + 1e-5f);
#pragma unroll
  for (int i = 0; i < 3; ++i) {
    const int c = t + i * 128;
    if (c < DIMP)
      hr[c] = (c < DIM) ? (x[i] - mu) * rstd * gamma[c] + beta[c] : 0.0f;
  }
}

// ---------------------------------------------------------------------------
// Host side
// ---------------------------------------------------------------------------
extern "C" void kernel_launch(void* const* d_in, const int* in_sizes, int n_in,
                              void* d_out, int out_size, void* d_ws,
                              size_t ws_size, hipStream_t stream) {
  (void)in_sizes; (void)n_in; (void)out_size; (void)ws_size;
  const int* Ms = (const int*)d_in[0];
  const float* Ev = (const float*)d_in[1];
  const float* Ey = (const float*)d_in[2];
  const float* WA = (const float*)d_in[3];
  const float* WO = (const float*)d_in[4];
  const float* gamma = (const float*)d_in[5];
  const float* beta = (const float*)d_in[6];

  float* out = (float*)d_out;
  float* logits = out;                                   // [4096, 30000]
  float* Aout = out + (long long)ROWS * YDIM;            // [8, 512, 512]

  float* ws = (float*)d_ws;
  float* X = ws;                                         // [4096, 304]
  float* XW = X + (long long)ROWS * DIMP;                // [4096, 304] (reused as H)
  float* Sr = XW + (long long)ROWS * DIMP;               // [8, 512, 512]
  float* AX = Sr + (long long)BATCH * SEQ * SEQ;         // [4096, 304]
  float* Qc = AX + (long long)ROWS * DIMP;               // [4096, 304]

  // 1) X = E_v[M_s]
  gather_embed<<<ROWS, 128, 0, stream>>>(Ms, Ev, X);

  // 2) XW = X @ W_A          (M=4096, N=300, K=300)
  {
    dim3 g((DIM + 127) / 128, ROWS / 128, 1);
    gemm_wmma_f32<false><<<g, 256, 0, stream>>>(X, WA, XW, ROWS, DIM, DIM,
                                                DIMP, DIM, DIMP, 0, 0, 0);
  }
  // 3) S = XW_b @ X_b^T      (per batch, M=N=512, K=300)
  {
    dim3 g(SEQ / 128, SEQ / 128, BATCH);
    gemm_wmma_f32<true><<<g, 256, 0, stream>>>(
        XW, X, Sr, SEQ, SEQ, DIM, DIMP, DIMP, SEQ,
        (long long)SEQ * DIMP, (long long)SEQ * DIMP, (long long)SEQ * SEQ);
  }
  // 4) A = softmax(mask(S))  -> written straight into the A output buffer
  mask_softmax<<<ROWS, 256, 0, stream>>>(Sr, Aout);

  // 5) AX = A_b @ X_b        (per batch, M=512, N=300, K=512)
  {
    dim3 g((DIM + 127) / 128, SEQ / 128, BATCH);
    gemm_wmma_f32<false><<<g, 256, 0, stream>>>(
        Aout, X, AX, SEQ, DIM, SEQ, SEQ, DIMP, DIMP,
        (long long)SEQ * SEQ, (long long)SEQ * DIMP, (long long)SEQ * DIMP);
  }
  // 6) Qc = AX @ W_O         (M=4096, N=300, K=300)
  {
    dim3 g((DIM + 127) / 128, ROWS / 128, 1);
    gemm_wmma_f32<false><<<g, 256, 0, stream>>>(AX, WO, Qc, ROWS, DIM, DIM,
                                                DIMP, DIM, DIMP, 0, 0, 0);
  }
  // 7) H = LN(Qc + X)  (H reuses XW buffer)
  residual_layernorm<<<ROWS, 128, 0, stream>>>(Qc, X, gamma, beta, XW);

  // 8) logits = H @ E_y^T    (M=4096, N=30000, K=300) — output-BW bound
  {
    dim3 g((YDIM + 127) / 128, ROWS / 128, 1);
    gemm_wmma_f32<true><<<g, 256, 0, stream>>>(XW, Ey, logits, ROWS, YDIM, DIM,
                                               DIMP, DIM, YDIM, 0, 0, 0);
  }
}